// VQ_36361193127971
// MI455X (gfx1250) — compile-verified
//
#include <hip/hip_runtime.h>

typedef float v2f  __attribute__((ext_vector_type(2)));
typedef float v8f  __attribute__((ext_vector_type(8)));
typedef float f32x4 __attribute__((ext_vector_type(4)));

// Problem constants (from reference)
constexpr int Bn = 32, Dd = 8, Hh = 128, Ww = 128;
constexpr int HW = Hh * Ww;            // 16384
constexpr int Np = Bn * HW;            // 524288 pixels
constexpr int Kc = 256;                // codes
// Main-kernel tiling
constexpr int WAVES = 8;               // waves per block (256 threads, wave32)
constexpr int TPW   = 8;               // 16-pixel tiles per wave
constexpr int PPB   = WAVES * TPW * 16;            // 1024 pixels / block
constexpr int GRID_MAIN = Np / PPB;                // 512 blocks (exact)
// d_out layout (floats), outputs concatenated in return order
constexpr size_t EMB_OFF  = 0;                         // [B,D,H,W]
constexpr size_t Y_OFF    = (size_t)Np * Dd;           // [B,H,W]
constexpr size_t REP_OFF  = Y_OFF + (size_t)Np;        // [B,H,W,K]
constexpr size_t LOSS_OFF = REP_OFF + (size_t)Np * Kc; // loss, Hy, Hyx

__global__ void vq_init_ws(unsigned* __restrict__ hist) {
  hist[threadIdx.x] = 0u;  // 1 block, 256 threads
}

__global__ __launch_bounds__(256) void vq_main(
    const float* __restrict__ q, const float* __restrict__ cbg,
    float* __restrict__ out, unsigned* __restrict__ hist_g,
    float* __restrict__ blockLoss) {
  __shared__ float    cb[Kc * Dd];   // codebook, row-major [k][d]
  __shared__ float    c2s[Kc];       // ||c_k||^2
  __shared__ unsigned histS[Kc];
  __shared__ float    wsum[WAVES];

  const int tid = threadIdx.x;
  for (int i = tid; i < Kc * Dd; i += 256) cb[i] = cbg[i];
  __syncthreads();
  if (tid < Kc) {
    float s = 0.f;
#pragma unroll
    for (int d = 0; d < Dd; ++d) { float c = cb[tid * Dd + d]; s += c * c; }
    c2s[tid] = s;
    histS[tid] = 0u;
  }
  __syncthreads();

  const int lane = tid & 31;
  const int wave = tid >> 5;
  const int nsub = lane & 15;   // column-in-tile (B/C/D) and row m (A)
  const int hi   = lane >> 4;   // lane half
  const int kb   = 2 * hi;      // per-lane k base within a k=4 fragment

  // Preload B fragments (codebook^T) and bias = -0.5*||c||^2 into registers.
  // B layout (f32 4x16): VGPR v, lane l -> k = 2*(l/16)+v, n = l%16.
  v2f   Bf[16][2];
  float bias[16];
#pragma unroll
  for (int ct = 0; ct < 16; ++ct) {
    const float* row = &cb[(ct * 16 + nsub) * Dd];
    Bf[ct][0] = v2f{row[kb],     row[kb + 1]};      // dims 0..3 fragment
    Bf[ct][1] = v2f{row[4 + kb], row[5 + kb]};      // dims 4..7 fragment
    bias[ct]  = -0.5f * c2s[ct * 16 + nsub];
  }

  float lossAcc = 0.f;
  float* yout = out + Y_OFF;

  for (int pt = 0; pt < TPW; ++pt) {
    const int gt = (blockIdx.x * WAVES + wave) * TPW + pt;
    const int p0 = gt * 16;
    // A layout (f32 16x4): lane l -> m = l%16, VGPR v -> k = 2*(l/16)+v
    const unsigned p    = (unsigned)(p0 + nsub);
    const unsigned bimg = p >> 14;            // / HW
    const unsigned hw   = p & (HW - 1);
    const float* qb = q + (size_t)bimg * Dd * HW + hw;
    v2f a0 = v2f{qb[(size_t)kb * HW],       qb[(size_t)(kb + 1) * HW]};
    v2f a1 = v2f{qb[(size_t)(4 + kb) * HW], qb[(size_t)(5 + kb) * HW]};
    float x2p = a0.x * a0.x + a0.y * a0.y + a1.x * a1.x + a1.y * a1.y;
    x2p += __shfl_xor(x2p, 16, 32);           // full ||x||^2, held per pixel l%16

    float bestv[8];
    int   besti[8];
#pragma unroll
    for (int r = 0; r < 8; ++r) { bestv[r] = -3.4e38f; besti[r] = 0; }

#pragma unroll
    for (int ct = 0; ct < 16; ++ct) {
      v8f acc;
#pragma unroll
      for (int r = 0; r < 8; ++r) acc[r] = bias[ct];
      // D = A(16x4) * B(4x16) + C ; chain two fragments for the full D=8 dot
      acc = __builtin_amdgcn_wmma_f32_16x16x4_f32(
          false, a0, false, Bf[ct][0], (short)0, acc, false, false);
      acc = __builtin_amdgcn_wmma_f32_16x16x4_f32(
          false, a1, false, Bf[ct][1], (short)0, acc, false, false);
      const int nidx = ct * 16 + nsub;        // code index this lane holds
#pragma unroll
      for (int r = 0; r < 8; ++r) {
        float v = acc[r];
        if (v > bestv[r]) { bestv[r] = v; besti[r] = nidx; }
      }
    }
    // Cross-lane argmax within each 16-lane half (ties -> lower index)
#pragma unroll
    for (int s = 1; s < 16; s <<= 1) {
#pragma unroll
      for (int r = 0; r < 8; ++r) {
        float ov = __shfl_xor(bestv[r], s, 32);
        int   oi = __shfl_xor(besti[r], s, 32);
        if (ov > bestv[r] || (ov == bestv[r] && oi < besti[r])) {
          bestv[r] = ov; besti[r] = oi;
        }
      }
    }
    // Lanes 0..7 own rows 0..7, lanes 16..23 own rows 8..15
    const int rsel = lane & 7;
    float bv = bestv[0]; int bi = besti[0];
#pragma unroll
    for (int r = 1; r < 8; ++r)
      if (rsel == r) { bv = bestv[r]; bi = besti[r]; }
    const int m   = hi * 8 + rsel;
    const float x2m = __shfl(x2p, m, 32);     // pixel m's ||x||^2 from lane m
    if (nsub < 8) {
      const unsigned pm   = (unsigned)(p0 + m);
      const unsigned bm   = pm >> 14;
      const unsigned hwm  = pm & (HW - 1);
      yout[pm] = (float)bi;
      float* eb = out + EMB_OFF + (size_t)bm * Dd * HW + hwm;
#pragma unroll
      for (int d = 0; d < Dd; ++d) eb[(size_t)d * HW] = cb[bi * Dd + d];
      // sum_d (q - x)^2 = x2 - 2*(x.c - 0.5*c2) = x2 - 2*bestscore
      lossAcc += x2m - 2.f * bv;
      atomicAdd(&histS[bi], 1u);
    }
  }

  // Deterministic loss reduction: shuffle tree -> per-wave -> per-block slot
#pragma unroll
  for (int s = 16; s >= 1; s >>= 1) lossAcc += __shfl_xor(lossAcc, s, 32);
  if (lane == 0) wsum[wave] = lossAcc;
  __syncthreads();
  if (tid == 0) {
    float t = 0.f;
    for (int w = 0; w < WAVES; ++w) t += wsum[w];
    blockLoss[blockIdx.x] = t;
  }
  if (tid < Kc) atomicAdd(&hist_g[tid], histS[tid]);
}

// 512 MB one-hot writer: 8 threads/pixel, 8x 128-bit NT stores each.
__global__ __launch_bounds__(256) void vq_onehot(
    const float* __restrict__ yout, float* __restrict__ rep) {
  const unsigned gid = blockIdx.x * 256u + threadIdx.x;
  const unsigned pix = gid >> 3;
  const unsigned j   = gid & 7u;
  const int y = (int)yout[pix];
  float* base = rep + (size_t)pix * Kc;
#pragma unroll
  for (int qq = 0; qq < 8; ++qq) {
    const int k0 = qq * 32 + (int)j * 4;
    f32x4 v;
    v.x = (y == k0)     ? 1.f : 0.f;
    v.y = (y == k0 + 1) ? 1.f : 0.f;
    v.z = (y == k0 + 2) ? 1.f : 0.f;
    v.w = (y == k0 + 3) ? 1.f : 0.f;
    __builtin_nontemporal_store(v, (f32x4*)(base + k0));
  }
}

__global__ __launch_bounds__(256) void vq_finalize(
    const unsigned* __restrict__ hist, const float* __restrict__ blockLoss,
    float* __restrict__ out) {
  __shared__ float wsum[8];
  const int tid  = threadIdx.x;
  const int lane = tid & 31;
  const int wave = tid >> 5;
  float py = (float)hist[tid] * (1.0f / (float)Np);
  float e  = -py * __log2f(py + 1e-10f);
#pragma unroll
  for (int s = 16; s >= 1; s >>= 1) e += __shfl_xor(e, s, 32);
  if (lane == 0) wsum[wave] = e;
  __syncthreads();
  if (tid == 0) {
    float Hy = 0.f;
    for (int w = 0; w < 8; ++w) Hy += wsum[w];
    float S = 0.f;
    for (int b = 0; b < GRID_MAIN; ++b) S += blockLoss[b];  // fixed order
    // loss = 0.5*(BETA*S + S)/(B*H)/TAU with BETA=0.25, TAU=1, B*H=4096
    out[LOSS_OFF]     = 0.625f * S * (1.0f / 4096.0f);
    out[LOSS_OFF + 1] = Hy;
    out[LOSS_OFF + 2] = 0.f;   // Hyx: entropy of one-hot == 0
  }
}

extern "C" void kernel_launch(void* const* d_in, const int* in_sizes, int n_in,
                              void* d_out, int out_size, void* d_ws, size_t ws_size,
                              hipStream_t stream) {
  const float* q   = (const float*)d_in[0];   // query  [32,8,128,128] f32
  const float* cbg = (const float*)d_in[1];   // codebook [256,8] f32
  float* out = (float*)d_out;
  unsigned* hist   = (unsigned*)d_ws;                            // 256 u32
  float* blockLoss = (float*)((char*)d_ws + Kc * sizeof(unsigned)); // 512 f32

  vq_init_ws<<<1, 256, 0, stream>>>(hist);
  vq_main<<<GRID_MAIN, 256, 0, stream>>>(q, cbg, out, hist, blockLoss);
  vq_onehot<<<(Np * 8) / 256, 256, 0, stream>>>(out + Y_OFF, out + REP_OFF);
  vq_finalize<<<1, 256, 0, stream>>>(hist, blockLoss, out);
}